// Yolov2Loss_90975997263957
// MI455X (gfx1250) — compile-verified
//
#include <hip/hip_runtime.h>

#define LAMBDA_COORD 5.0f
#define LAMBDA_NOOBJ 0.5f

constexpr int BB = 64;   // batch
constexpr int SS = 64;   // grid cells per side
constexpr int AA = 5;    // anchors
constexpr int CC = 20;   // classes
constexpr int TT = 20;   // targets per image
constexpr int PRED_STRIDE = AA * (5 + CC);      // 125 floats per cell
constexpr int NCELLS = BB * SS * SS;            // 262144 cells
constexpr int P1_BLOCK = 256;
constexpr int P1_BLOCKS = NCELLS / P1_BLOCK;    // 1024 blocks, 1 cell/thread
constexpr int NPART = P1_BLOCKS + 1;            // part1 partials + part2 partial

typedef __attribute__((ext_vector_type(2))) float v2f;
typedef __attribute__((ext_vector_type(8))) float v8f;

// Fast sigmoid: v_exp_f32 + v_rcp_f32 (~1 ulp), no precise-div fixup sequence.
__device__ __forceinline__ float sigmoidf_(float x) {
    return __builtin_amdgcn_rcpf(1.0f + __expf(-x));
}

// Exact wave32 sum using V_WMMA_F32_16X16X4_F32 against an all-ones B matrix.
// A layout (16x4 f32): VGPR0 lane L<16 -> A[L][0], lane L>=16 -> A[L-16][2];
// VGPR1 left zero. With B = ones, D[m][n] = A[m][0] + A[m][2] = s_m + s_{m+16}.
// D layout: VGPR r lanes<16 hold row r, lanes>=16 hold row r+8. Summing the 8
// D VGPRs per lane and folding the two lane halves gives the full 32-lane sum.
__device__ __forceinline__ float wave_sum_wmma(float v) {
    v2f a; a.x = v;    a.y = 0.0f;
    v2f b; b.x = 1.0f; b.y = 1.0f;
    v8f c = {};
    c = __builtin_amdgcn_wmma_f32_16x16x4_f32(false, a, false, b, (short)0, c,
                                              false, false);
    float t = ((c[0] + c[1]) + (c[2] + c[3])) + ((c[4] + c[5]) + (c[6] + c[7]));
    t += __shfl_xor(t, 16, 32);
    return t;
}

// ---------------------------------------------------------------------------
// Pass 1: streaming reduction of sigmoid(tobj)^2 over all B*S*S cells x A
// anchors. One thread per cell: 5 independent tobj loads (offsets a*25+4),
// no loops, no integer division. HBM-bound: ~131 MB / 23.3 TB/s ~= 5.6 us.
// ---------------------------------------------------------------------------
__global__ void yolo_noobj_kernel(const float* __restrict__ pred,
                                  float* __restrict__ partials) {
    const int cell = blockIdx.x * P1_BLOCK + threadIdx.x;   // always < NCELLS
    const float* cp = pred + (size_t)cell * PRED_STRIDE;

    float acc = 0.0f;
    #pragma unroll
    for (int a = 0; a < AA; ++a) {
        float conf = sigmoidf_(cp[a * (5 + CC) + 4]);
        acc += conf * conf;
    }

    // All 256 threads reach here (no divergence) -> EXEC all-ones for WMMA.
    float wsum = wave_sum_wmma(acc);

    __shared__ float wpart[8];
    int lane = threadIdx.x & 31;
    int wave = threadIdx.x >> 5;
    if (lane == 0) wpart[wave] = wsum;
    __syncthreads();
    if (threadIdx.x == 0) {
        float s = 0.0f;
        #pragma unroll
        for (int w = 0; w < 8; ++w) s += wpart[w];
        partials[blockIdx.x] = LAMBDA_NOOBJ * s;
    }
}

// ---------------------------------------------------------------------------
// Pass 2: per-target obj/coord terms + correction of the no-obj double count.
// Single block; deterministic last-target-wins dedup (mirrors JAX scatter-set).
// ---------------------------------------------------------------------------
__global__ void yolo_obj_kernel(const float* __restrict__ pred,
                                const float* __restrict__ tgt,
                                const float* __restrict__ anchors,
                                float* __restrict__ partial_out) {
    __shared__ int   keys[BB * TT];
    __shared__ float wpart[8];

    float aw[AA], ah[AA];
    #pragma unroll
    for (int a = 0; a < AA; ++a) { aw[a] = anchors[2 * a]; ah[a] = anchors[2 * a + 1]; }

    // Pass 2a: cell+anchor key per target (argmax wh-IoU, first max wins).
    for (int t = threadIdx.x; t < BB * TT; t += blockDim.x) {
        const float* tp = tgt + t * 5;
        float gx = tp[0], gy = tp[1], gw = tp[2], gh = tp[3];
        int gi = min(max((int)(gx * (float)SS), 0), SS - 1);
        int gj = min(max((int)(gy * (float)SS), 0), SS - 1);
        int best = 0; float bestv = -1.0f;
        #pragma unroll
        for (int a = 0; a < AA; ++a) {
            float inter = fminf(gw, aw[a]) * fminf(gh, ah[a]);
            float uni   = gw * gh + aw[a] * ah[a] - inter;
            float iou   = uni > 0.0f ? inter / uni : 0.0f;
            if (iou > bestv) { bestv = iou; best = a; }
        }
        int b = t / TT;
        keys[t] = (((b * AA + best) * SS) + gj) * SS + gi;
    }
    __syncthreads();

    // Pass 2b: winners (no later target in same image hits same key) contribute.
    float acc = 0.0f;
    for (int t = threadIdx.x; t < BB * TT; t += blockDim.x) {
        int b = t / TT, s = t - b * TT;
        int key = keys[t];
        bool win = true;
        for (int s2 = s + 1; s2 < TT; ++s2)
            if (keys[b * TT + s2] == key) { win = false; break; }
        if (!win) continue;

        int gi = key & (SS - 1);
        int gj = (key >> 6) & (SS - 1);
        int ba = (key >> 12) - b * AA;

        const float* tp = tgt + t * 5;
        float gx = tp[0], gy = tp[1], gw = tp[2], gh = tp[3];

        const float* pp = pred + (size_t)((b * SS + gj) * SS + gi) * PRED_STRIDE
                        + ba * (5 + CC);
        float bx   = sigmoidf_(pp[0]);
        float by   = sigmoidf_(pp[1]);
        float bw   = __expf(pp[2]) * aw[ba];
        float bh   = __expf(pp[3]) * ah[ba];
        float conf = sigmoidf_(pp[4]);

        float txo = gx * (float)SS - (float)gi;
        float tyo = gy * (float)SS - (float)gj;
        const float eps = 1e-6f;
        float dcx = bx - txo, dcy = by - tyo;
        float dsw = sqrtf(bw + eps) - sqrtf(gw + eps);
        float dsh = sqrtf(bh + eps) - sqrtf(gh + eps);

        float gws = gw * (float)SS, ghs = gh * (float)SS;
        float inter = fminf(bw, gws) * fminf(bh, ghs);
        float uni   = bw * bh + gws * ghs - inter;
        float iou   = uni > 0.0f ? inter / uni : 0.0f;
        float d     = iou - conf;

        acc += LAMBDA_COORD * (dcx * dcx + dcy * dcy)
             + LAMBDA_COORD * (dsw * dsw + dsh * dsh)
             + d * d
             - LAMBDA_NOOBJ * conf * conf;   // undo pass-1 no-obj count at obj cell
    }

    #pragma unroll
    for (int m = 16; m >= 1; m >>= 1) acc += __shfl_xor(acc, m, 32);
    int lane = threadIdx.x & 31, wave = threadIdx.x >> 5;
    if (lane == 0) wpart[wave] = acc;
    __syncthreads();
    if (threadIdx.x == 0) {
        float s = 0.0f;
        #pragma unroll
        for (int w = 0; w < 8; ++w) s += wpart[w];
        partial_out[0] = s;
    }
}

// ---------------------------------------------------------------------------
// Pass 3: deterministic finalize of all partials -> scalar loss.
// ---------------------------------------------------------------------------
__global__ void yolo_finalize_kernel(const float* __restrict__ partials,
                                     float* __restrict__ out) {
    float acc = 0.0f;
    for (int i = threadIdx.x; i < NPART; i += blockDim.x) acc += partials[i];
    #pragma unroll
    for (int m = 16; m >= 1; m >>= 1) acc += __shfl_xor(acc, m, 32);
    __shared__ float wpart[8];
    int lane = threadIdx.x & 31, wave = threadIdx.x >> 5;
    if (lane == 0) wpart[wave] = acc;
    __syncthreads();
    if (threadIdx.x == 0) {
        float s = 0.0f;
        #pragma unroll
        for (int w = 0; w < 8; ++w) s += wpart[w];
        out[0] = s;
    }
}

extern "C" void kernel_launch(void* const* d_in, const int* in_sizes, int n_in,
                              void* d_out, int out_size, void* d_ws, size_t ws_size,
                              hipStream_t stream) {
    const float* pred    = (const float*)d_in[0];
    const float* tgt     = (const float*)d_in[1];
    const float* anchors = (const float*)d_in[2];
    float* out      = (float*)d_out;
    float* partials = (float*)d_ws;   // NPART floats of scratch

    yolo_noobj_kernel<<<P1_BLOCKS, P1_BLOCK, 0, stream>>>(pred, partials);
    yolo_obj_kernel<<<1, 256, 0, stream>>>(pred, tgt, anchors, partials + P1_BLOCKS);
    yolo_finalize_kernel<<<1, 256, 0, stream>>>(partials, out);
}